// OperationCell_66554813218862
// MI455X (gfx1250) — compile-verified
//
#include <hip/hip_runtime.h>
#include <hip/hip_bf16.h>
#include <math.h>
#include <stdint.h>

// LSTM cell, D=4096.  Memory-bound: 512 MB of fp32 weights streamed once
// (~22 us floor at 23.3 TB/s).
// Kernel 1: 4 GEMVs via V_WMMA_F32_16X16X4_F32 (broadcast-A), weight tiles
//           streamed into double-buffered LDS by the Tensor Data Mover
//           (tensor_load_to_lds, TENSORcnt), split-K partials into d_ws.
// Kernel 2: reduce splits + bias + sigmoid/tanh elementwise combine.

#define DDIM   4096
#define TWO_D  8192
#define BN     128          // output columns per block (8 waves x 16)
#define KTILE  64           // rows of W staged in LDS per tile
#define NSPLIT 8            // split-K factor
#define KC     (TWO_D / NSPLIT)  // 1024 K-rows per block
#define LDSW   136          // padded LDS row stride (dwords): 128 + 8

typedef float v2f __attribute__((ext_vector_type(2)));
typedef float v8f __attribute__((ext_vector_type(8)));
typedef unsigned int u32x4 __attribute__((ext_vector_type(4)));
typedef int i32x4 __attribute__((ext_vector_type(4)));
typedef int i32x8 __attribute__((ext_vector_type(8)));

#define HAS_TDM __has_builtin(__builtin_amdgcn_tensor_load_to_lds)

#if HAS_TDM
// Issue one TDM DMA: 64x128 fp32 tile of W (row pitch DDIM) -> LDS at lds_off,
// with 8-dword padding after every 128-dword row (LDSW layout).
__device__ __forceinline__ void tdm_load_tile(const float* gsrc, uint32_t lds_off) {
    const uint64_t ga = (uint64_t)(uintptr_t)gsrc;
    u32x4 g0;
    g0.x = 1u;                                            // count=1, user D#
    g0.y = lds_off;                                       // lds_addr (bytes)
    g0.z = (uint32_t)ga;                                  // global_addr[31:0]
    g0.w = (uint32_t)((ga >> 32) & 0x01FFFFFFu) | (2u << 30); // addr[56:32], type=2
    i32x8 g1;
    g1[0] = (int)((2u << 16)     // data_size = 4 bytes
                | (1u << 20)     // pad_enable
                | (6u << 22)     // pad_interval: every 128 dwords
                | (7u << 25));   // pad_amount: 8 dwords
    g1[1] = (int)(((uint32_t)BN & 0xFFFFu) << 16);        // tensor_dim0[15:0]=128
    g1[2] = (int)(((uint32_t)KTILE & 0xFFFFu) << 16);     // tensor_dim1[15:0]=64
    g1[3] = (int)(((uint32_t)BN) << 16);                  // tile_dim0=128
    g1[4] = (int)((uint32_t)KTILE);                       // tile_dim1=64, tile_dim2=0
    g1[5] = (int)DDIM;                                    // tensor_dim0_stride=4096
    g1[6] = 0;
    g1[7] = 0;
    i32x4 z4 = {0, 0, 0, 0};
#if defined(__clang_major__) && (__clang_major__ >= 23)
    i32x8 z8 = {0, 0, 0, 0, 0, 0, 0, 0};
    __builtin_amdgcn_tensor_load_to_lds(g0, g1, z4, z4, z8, 0);
#else
    __builtin_amdgcn_tensor_load_to_lds(g0, g1, z4, z4, 0);
#endif
}
#endif

__global__ __launch_bounds__(256) void lstm_gates_wmma(
    const float* __restrict__ x, const float* __restrict__ h,
    const float* __restrict__ Wf, const float* __restrict__ Wi,
    const float* __restrict__ Wc, const float* __restrict__ Wo,
    float* __restrict__ part)
{
    __shared__ float sW[2][KTILE][LDSW];   // 2 x 34816 B (double buffer)
    __shared__ float sS[KC];               // 4096 B  (concat chunk)

    const int jt    = blockIdx.x;       // 0..31  column tile
    const int split = blockIdx.y;       // 0..7   K split
    const int g     = blockIdx.z;       // 0..3   gate
    const float* W  = (g == 0) ? Wf : (g == 1) ? Wi : (g == 2) ? Wc : Wo;

    const int tid     = threadIdx.x;
    const int wave    = tid >> 5;       // 0..7
    const int lane    = tid & 31;
    const int halfsel = (lane >= 16) ? 2 : 0;  // lanes 16-31 supply K=2,3
    const int col     = lane & 15;             // output column within wave tile

    const int j0 = jt * BN;
    const int k0 = split * KC;

    // Stage concat[k0 .. k0+KC) = [h ; x] chunk into LDS (coalesced).
    for (int i2 = tid; i2 < KC; i2 += 256) {
        const int k = k0 + i2;
        sS[i2] = (k < DDIM) ? h[k] : x[k - DDIM];
    }

    v8f acc0 = {};
    v8f acc1 = {};

    const int NT = KC / KTILE;          // 16 tiles
    const float* wbase = W + (size_t)k0 * DDIM + j0;

#if HAS_TDM
    const uint32_t lds_off0 = (uint32_t)(uintptr_t)&sW[0][0][0];
    const uint32_t lds_off1 = (uint32_t)(uintptr_t)&sW[1][0][0];

    if (wave == 0) {
        tdm_load_tile(wbase, lds_off0);                    // prefetch tile 0
    }
    for (int t = 0; t < NT; ++t) {
        if (wave == 0) {
            if (t + 1 < NT) {
                tdm_load_tile(wbase + (size_t)(t + 1) * KTILE * DDIM,
                              ((t + 1) & 1) ? lds_off1 : lds_off0);
                __builtin_amdgcn_s_wait_tensorcnt(1);      // tile t has landed
            } else {
                __builtin_amdgcn_s_wait_tensorcnt(0);      // last tile landed
            }
        }
        __syncthreads();                                   // tile t visible to all
        const float (*buf)[LDSW] = sW[t & 1];
        const int kbase = t * KTILE;
        #pragma unroll
        for (int ks = 0; ks < KTILE; ks += 8) {
            {
                const v2f A = *(const v2f*)(&sS[kbase + ks + halfsel]);
                v2f B;
                B.x = buf[ks + halfsel][wave * 16 + col];
                B.y = buf[ks + halfsel + 1][wave * 16 + col];
                acc0 = __builtin_amdgcn_wmma_f32_16x16x4_f32(
                    false, A, false, B, (short)0, acc0, false, false);
            }
            {
                const v2f A = *(const v2f*)(&sS[kbase + ks + 4 + halfsel]);
                v2f B;
                B.x = buf[ks + 4 + halfsel][wave * 16 + col];
                B.y = buf[ks + 5 + halfsel][wave * 16 + col];
                acc1 = __builtin_amdgcn_wmma_f32_16x16x4_f32(
                    false, A, false, B, (short)0, acc1, false, false);
            }
        }
        __syncthreads();   // all reads of buf done before tile t+2 DMA overwrites it
    }
#else
    // Fallback: manual staging through VGPRs into sW[0] (single buffer).
    for (int t = 0; t < NT; ++t) {
        __syncthreads();
        const float* gsrc = wbase + (size_t)t * KTILE * DDIM;
        #pragma unroll
        for (int rr = 0; rr < KTILE / 8; ++rr) {
            const int r = wave + rr * 8;
            const float4 v = *(const float4*)(gsrc + (size_t)r * DDIM + (lane << 2));
            *(float4*)(&sW[0][r][lane << 2]) = v;
        }
        if (t + 1 < NT) {
            __builtin_prefetch(gsrc + (size_t)(KTILE + (tid >> 2)) * DDIM + ((tid & 3) << 5), 0, 0);
        }
        __syncthreads();
        const int kbase = t * KTILE;
        #pragma unroll
        for (int ks = 0; ks < KTILE; ks += 8) {
            {
                const v2f A = *(const v2f*)(&sS[kbase + ks + halfsel]);
                v2f B;
                B.x = sW[0][ks + halfsel][wave * 16 + col];
                B.y = sW[0][ks + halfsel + 1][wave * 16 + col];
                acc0 = __builtin_amdgcn_wmma_f32_16x16x4_f32(
                    false, A, false, B, (short)0, acc0, false, false);
            }
            {
                const v2f A = *(const v2f*)(&sS[kbase + ks + 4 + halfsel]);
                v2f B;
                B.x = sW[0][ks + 4 + halfsel][wave * 16 + col];
                B.y = sW[0][ks + 5 + halfsel][wave * 16 + col];
                acc1 = __builtin_amdgcn_wmma_f32_16x16x4_f32(
                    false, A, false, B, (short)0, acc1, false, false);
            }
        }
    }
#endif

    // Row 0 of D lives in acc[0] lanes 0..15 (all D rows identical since all
    // A rows are identical).  One partial per (gate, split, column).
    if (lane < 16) {
        part[(size_t)(g * NSPLIT + split) * DDIM + j0 + wave * 16 + col] =
            acc0[0] + acc1[0];
    }
}

__global__ __launch_bounds__(256) void lstm_combine(
    const float* __restrict__ c,
    const float* __restrict__ bf, const float* __restrict__ bi,
    const float* __restrict__ bc, const float* __restrict__ bo,
    const float* __restrict__ part, float* __restrict__ out)
{
    const int j = blockIdx.x * blockDim.x + threadIdx.x;
    if (j >= DDIM) return;
    float f  = bf[j];
    float ig = bi[j];
    float cg = bc[j];
    float og = bo[j];
    #pragma unroll
    for (int s = 0; s < NSPLIT; ++s) {
        f  += part[(size_t)(0 * NSPLIT + s) * DDIM + j];
        ig += part[(size_t)(1 * NSPLIT + s) * DDIM + j];
        cg += part[(size_t)(2 * NSPLIT + s) * DDIM + j];
        og += part[(size_t)(3 * NSPLIT + s) * DDIM + j];
    }
    const float zf = 1.0f / (1.0f + __expf(-f));
    const float zi = 1.0f / (1.0f + __expf(-ig));
    const float zo = 1.0f / (1.0f + __expf(-og));
    const float z  = tanhf(cg);
    const float cn = c[j] * zf + z * zi;
    out[j] = zo * tanhf(cn);
}

extern "C" void kernel_launch(void* const* d_in, const int* in_sizes, int n_in,
                              void* d_out, int out_size, void* d_ws, size_t ws_size,
                              hipStream_t stream) {
    (void)in_sizes; (void)n_in; (void)out_size; (void)ws_size;
    const float* x  = (const float*)d_in[0];
    const float* c  = (const float*)d_in[1];
    const float* h  = (const float*)d_in[2];
    const float* Wf = (const float*)d_in[3];
    const float* bf = (const float*)d_in[4];
    const float* Wi = (const float*)d_in[5];
    const float* bi = (const float*)d_in[6];
    const float* Wc = (const float*)d_in[7];
    const float* bc = (const float*)d_in[8];
    const float* Wo = (const float*)d_in[9];
    const float* bo = (const float*)d_in[10];
    float* part = (float*)d_ws;           // 4 * NSPLIT * 4096 fp32 = 512 KB
    float* out  = (float*)d_out;

    dim3 grid1(DDIM / BN, NSPLIT, 4);     // 32 x 8 x 4 = 1024 blocks
    lstm_gates_wmma<<<grid1, 256, 0, stream>>>(x, h, Wf, Wi, Wc, Wo, part);
    lstm_combine<<<(DDIM + 255) / 256, 256, 0, stream>>>(c, bf, bi, bc, bo, part, out);
}